// BasicLayer_14620068675725
// MI455X (gfx1250) — compile-verified
//
#include <hip/hip_runtime.h>
#include <hip/hip_bf16.h>
#include <stdint.h>

// Problem constants (match reference)
#define B_       4
#define H_       96
#define W_       96
#define C_       256
#define NBHD_    48
#define T_       3025
#define OUTD_    512
#define N_       (H_*W_)       // 9216
#define INNER_   4
#define KEEP_    2304          // N * 0.25
#define SAMPLE_  1728          // KEEP - RESERVE
#define RESERVE_ 576
#define KDIM_    (INNER_*C_)   // 1024

typedef __attribute__((ext_vector_type(2))) float v2f;
typedef __attribute__((ext_vector_type(8))) float v8f;

__device__ __forceinline__ float gelu_tanh(float x) {
  // jax.nn.gelu default (approximate=True)
  const float k0 = 0.7978845608028654f;  // sqrt(2/pi)
  const float k1 = 0.044715f;
  float x3 = x * x * x;
  return 0.5f * x * (1.0f + tanhf(k0 * (x + k1 * x3)));
}

// ---------------------------------------------------------------------------
// Kernel 1: wt = gelu(LN(pre_table @ w1 + b1))   -> [T, 4]
// ---------------------------------------------------------------------------
__global__ void wt_table_kernel(const float* __restrict__ pre_table,
                                const float* __restrict__ w1,
                                const float* __restrict__ b1,
                                const float* __restrict__ ln1_w,
                                const float* __restrict__ ln1_b,
                                float* __restrict__ wt) {
  int t = blockIdx.x * blockDim.x + threadIdx.x;
  if (t >= T_) return;
  float p[5];
#pragma unroll
  for (int i = 0; i < 5; ++i) p[i] = pre_table[t * 5 + i];
  float y[INNER_];
#pragma unroll
  for (int m = 0; m < INNER_; ++m) {
    float acc = b1[m];
#pragma unroll
    for (int i = 0; i < 5; ++i) acc += p[i] * w1[i * INNER_ + m];
    y[m] = acc;
  }
  float mu = 0.25f * (y[0] + y[1] + y[2] + y[3]);
  float var = 0.f;
#pragma unroll
  for (int m = 0; m < INNER_; ++m) { float d = y[m] - mu; var += d * d; }
  var *= 0.25f;
  float inv = rsqrtf(var + 1e-5f);
#pragma unroll
  for (int m = 0; m < INNER_; ++m) {
    float v = (y[m] - mu) * inv * ln1_w[m] + ln1_b[m];
    wt[t * INNER_ + m] = gelu_tanh(v);
  }
}

// ---------------------------------------------------------------------------
// Kernel 2: per-batch exact top-k via full bitonic sort of 16384 packed keys
// in 128KB dynamic LDS (CDNA5: up to 320KB/WGP). Key = mapped(prob) << 32 |
// (0xFFFFFFFF - idx): sorting descending gives prob-desc, idx-asc tie-break,
// matching jax.lax.top_k. Also writes pos_down (output 0) and idxbuf.
// ---------------------------------------------------------------------------
__global__ __launch_bounds__(512) void topk_kernel(
    const float* __restrict__ learned_prob,
    int* __restrict__ idxbuf,
    float* __restrict__ pos_out) {
  extern __shared__ unsigned long long keys[];
  const int SZ = 16384;  // next pow2 >= N_
  const int b = blockIdx.x;
  const int tid = threadIdx.x;

  for (int i = tid; i < SZ; i += 512) {
    unsigned long long key = 0ull;
    if (i < N_) {
      float p = learned_prob[(size_t)b * N_ + i];
      int y = i / W_, x = i % W_;
      if (((y & 3) == 0) && ((x & 3) == 0)) p -= 100.0f;  // reserved -> never sampled
      unsigned int u = __float_as_uint(p);
      u = (u & 0x80000000u) ? ~u : (u | 0x80000000u);      // monotone float->uint
      key = ((unsigned long long)u << 32) |
            (unsigned long long)(0xFFFFFFFFu - (unsigned int)i);
    }
    keys[i] = key;
  }
  __syncthreads();

  // Bitonic sort, descending final order
  for (int k = 2; k <= SZ; k <<= 1) {
    for (int j = k >> 1; j > 0; j >>= 1) {
      for (int i = tid; i < SZ; i += 512) {
        int ixj = i ^ j;
        if (ixj > i) {
          unsigned long long a = keys[i], c = keys[ixj];
          bool descRegion = ((i & k) == 0);
          bool doSwap = descRegion ? (a < c) : (a > c);
          if (doSwap) { keys[i] = c; keys[ixj] = a; }
        }
      }
      __syncthreads();
    }
  }

  // Sampled tokens: top SAMPLE_ keys (prob-descending)
  for (int s = tid; s < SAMPLE_; s += 512) {
    unsigned int orig = 0xFFFFFFFFu - (unsigned int)(keys[s] & 0xFFFFFFFFull);
    int slot = b * KEEP_ + s;
    idxbuf[slot] = (int)orig;
    pos_out[(size_t)slot * 2 + 0] = (float)(orig / W_);
    pos_out[(size_t)slot * 2 + 1] = (float)(orig % W_);
  }
  // Reserved tokens in ascending index order: y,x multiples of 4
  for (int r = tid; r < RESERVE_; r += 512) {
    int y = 4 * (r / 24), x = 4 * (r % 24);
    int n = y * W_ + x;
    int slot = b * KEEP_ + SAMPLE_ + r;
    idxbuf[slot] = n;
    pos_out[(size_t)slot * 2 + 0] = (float)y;
    pos_out[(size_t)slot * 2 + 1] = (float)x;
  }
}

// ---------------------------------------------------------------------------
// Kernel 3: per kept token: gather 48 neighbors, weighted aggregate into a
// 4x256 = 1024-vector, LayerNorm -> Abuf row. One 256-thread block per token,
// thread = channel. 48 per-neighbor weight 4-vectors staged in LDS.
// Next-row global_prefetch issued to hide gather latency.
// ---------------------------------------------------------------------------
__global__ __launch_bounds__(256) void agg_kernel(
    const float* __restrict__ feat,
    const int*   __restrict__ member_idx,
    const float* __restrict__ cluster_mask,
    const float* __restrict__ learned_prob,
    const int*   __restrict__ pe_idx,
    const float* __restrict__ wt,
    const int*   __restrict__ idxbuf,
    const float* __restrict__ norm_w,
    const float* __restrict__ norm_b,
    float* __restrict__ Abuf) {
  __shared__ float sW[NBHD_][INNER_];
  __shared__ int   sM[NBHD_];
  __shared__ float redS[256];
  __shared__ float redQ[256];

  const int t   = blockIdx.x;       // 0 .. B*KEEP-1
  const int b   = t / KEEP_;
  const int tid = threadIdx.x;
  const int tok = idxbuf[t];

  if (tid < NBHD_) {
    size_t base = ((size_t)b * N_ + tok) * NBHD_ + tid;
    int   mj = member_idx[base];
    float cm = cluster_mask[base];
    int   pj = pe_idx[base];
    float lp = learned_prob[(size_t)b * N_ + mj];
    float s  = lp * cm;
    sM[tid] = mj;
#pragma unroll
    for (int m = 0; m < INNER_; ++m) sW[tid][m] = wt[pj * INNER_ + m] * s;
  }
  __syncthreads();

  const int c = tid;  // channel
  float acc0 = 0.f, acc1 = 0.f, acc2 = 0.f, acc3 = 0.f;
  const float* fb = feat + (size_t)b * N_ * C_;
  __builtin_prefetch(fb + (size_t)sM[0] * C_ + c, 0, 1);
#pragma unroll 4
  for (int j = 0; j < NBHD_; ++j) {
    if (j + 1 < NBHD_) __builtin_prefetch(fb + (size_t)sM[j + 1] * C_ + c, 0, 1);
    float f = fb[(size_t)sM[j] * C_ + c];   // coalesced 256-wide gather row
    acc0 += sW[j][0] * f;
    acc1 += sW[j][1] * f;
    acc2 += sW[j][2] * f;
    acc3 += sW[j][3] * f;
  }

  // Block LayerNorm over 1024 values (4 per thread)
  redS[tid] = acc0 + acc1 + acc2 + acc3;
  redQ[tid] = acc0*acc0 + acc1*acc1 + acc2*acc2 + acc3*acc3;
  __syncthreads();
  for (int off = 128; off > 0; off >>= 1) {
    if (tid < off) { redS[tid] += redS[tid + off]; redQ[tid] += redQ[tid + off]; }
    __syncthreads();
  }
  float mean = redS[0] * (1.0f / (float)KDIM_);
  float var  = redQ[0] * (1.0f / (float)KDIM_) - mean * mean;
  float inv  = rsqrtf(var + 1e-5f);

  float* arow = Abuf + (size_t)t * KDIM_;
  float av[INNER_] = {acc0, acc1, acc2, acc3};
#pragma unroll
  for (int m = 0; m < INNER_; ++m) {
    int e = m * C_ + c;   // einsum 'bnmc' layout: element = m*C + c
    arow[e] = (av[m] - mean) * inv * norm_w[e] + norm_b[e];
  }
}

// ---------------------------------------------------------------------------
// Kernel 4: fp32 GEMM [M=9216 x K=1024] @ [K x 512] + bias via
// V_WMMA_F32_16X16X4_F32 with 2x4 register tiling per wave:
// each wave owns a 32x64 output block (8 accumulators). Per K-step it loads
// 2 A fragments + 4 B fragments (1.5KB) feeding 8 WMMAs -> 192 B/wmma of L2
// traffic (vs 512 B/wmma untiled) and 8 independent WMMAs for the scheduler.
// A frag (16x4 f32): lanes 0-15 hold K=k0,k0+1; lanes 16-31 hold K=k0+2,k0+3.
// ---------------------------------------------------------------------------
__global__ __launch_bounds__(256) void gemm_kernel(
    const float* __restrict__ A,     // [M][KDIM_]
    const float* __restrict__ Wm,    // [KDIM_][OUTD_] row-major
    const float* __restrict__ bias,  // [OUTD_]
    float* __restrict__ Out) {       // [M][OUTD_]
  const int wavesN = OUTD_ / 64;     // 8 macro-columns of 64
  const int wave = (int)((blockIdx.x * blockDim.x + threadIdx.x) >> 5);
  const int lane = threadIdx.x & 31;
  const int wm = wave / wavesN;      // 0..287  (macro-row of 32)
  const int wn = wave % wavesN;      // 0..7    (macro-col of 64)

  const int half = lane >> 4;        // 0: K pair (0,1), 1: K pair (2,3)
  const int lid  = lane & 15;
  const int row0 = wm * 32 + lid;    // first A row this lane touches
  const int colb = wn * 64 + lid;    // base output column this lane touches

  const float* arp0 = A + (size_t)row0 * KDIM_;
  const float* arp1 = arp0 + (size_t)16 * KDIM_;

  v8f acc[2][4];
#pragma unroll
  for (int i = 0; i < 2; ++i)
#pragma unroll
    for (int t = 0; t < 4; ++t) acc[i][t] = (v8f){0.f,0.f,0.f,0.f,0.f,0.f,0.f,0.f};

#pragma unroll 2
  for (int k0 = 0; k0 < KDIM_; k0 += 4) {
    const int ka = k0 + half * 2;
    v2f a0 = *(const v2f*)(arp0 + ka);
    v2f a1 = *(const v2f*)(arp1 + ka);
    const float* bp0 = Wm + (size_t)ka * OUTD_ + colb;
    const float* bp1 = bp0 + OUTD_;
    v2f bv[4];
#pragma unroll
    for (int t = 0; t < 4; ++t) {      // columns colb + 16*t (64B imm offsets)
      bv[t].x = bp0[16 * t];
      bv[t].y = bp1[16 * t];
    }
#pragma unroll
    for (int t = 0; t < 4; ++t) {
      acc[0][t] = __builtin_amdgcn_wmma_f32_16x16x4_f32(
          false, a0, false, bv[t], (short)0, acc[0][t], false, false);
      acc[1][t] = __builtin_amdgcn_wmma_f32_16x16x4_f32(
          false, a1, false, bv[t], (short)0, acc[1][t], false, false);
    }
  }

#pragma unroll
  for (int t = 0; t < 4; ++t) {
    const int col = colb + 16 * t;
    const float bs = bias[col];
#pragma unroll
    for (int i = 0; i < 2; ++i) {
#pragma unroll
      for (int r = 0; r < 8; ++r) {
        const int row = wm * 32 + i * 16 + half * 8 + r;  // C/D layout halves
        Out[(size_t)row * OUTD_ + col] = acc[i][t][r] + bs;
      }
    }
  }
}

// ---------------------------------------------------------------------------
extern "C" void kernel_launch(void* const* d_in, const int* in_sizes, int n_in,
                              void* d_out, int out_size, void* d_ws, size_t ws_size,
                              hipStream_t stream) {
  // Inputs, setup_inputs() dict order:
  // 0 pos, 1 feat, 2 member_idx, 3 cluster_mask, 4 learned_prob, 5 pe_idx,
  // 6 pre_table, 7 w1, 8 b1, 9 ln1_w, 10 ln1_b, 11 norm_w, 12 norm_b,
  // 13 lin_w, 14 lin_b, 15 stride, 16 reserve_num
  const float* feat         = (const float*)d_in[1];
  const int*   member_idx   = (const int*)d_in[2];
  const float* cluster_mask = (const float*)d_in[3];
  const float* learned_prob = (const float*)d_in[4];
  const int*   pe_idx       = (const int*)d_in[5];
  const float* pre_table    = (const float*)d_in[6];
  const float* w1           = (const float*)d_in[7];
  const float* b1           = (const float*)d_in[8];
  const float* ln1_w        = (const float*)d_in[9];
  const float* ln1_b        = (const float*)d_in[10];
  const float* norm_w       = (const float*)d_in[11];
  const float* norm_b       = (const float*)d_in[12];
  const float* lin_w        = (const float*)d_in[13];
  const float* lin_b        = (const float*)d_in[14];

  // Outputs: pos_down [B,KEEP,2] then feat_down [B,KEEP,OUTD], concatenated flat
  float* pos_out  = (float*)d_out;
  float* feat_out = pos_out + (size_t)B_ * KEEP_ * 2;

  // Workspace layout
  float* wt     = (float*)d_ws;                           // T*4 floats
  int*   idxbuf = (int*)((char*)d_ws + 64 * 1024);        // B*KEEP ints
  float* Abuf   = (float*)((char*)d_ws + (1u << 20));     // M*KDIM floats (37.75 MB)

  // Allow 128KB dynamic LDS for the sort kernel (320KB/WGP on CDNA5)
  (void)hipFuncSetAttribute((const void*)topk_kernel,
                            hipFuncAttributeMaxDynamicSharedMemorySize,
                            16384 * (int)sizeof(unsigned long long));

  wt_table_kernel<<<(T_ + 255) / 256, 256, 0, stream>>>(
      pre_table, w1, b1, ln1_w, ln1_b, wt);

  topk_kernel<<<B_, 512, 16384 * sizeof(unsigned long long), stream>>>(
      learned_prob, idxbuf, pos_out);

  agg_kernel<<<B_ * KEEP_, 256, 0, stream>>>(
      feat, member_idx, cluster_mask, learned_prob, pe_idx,
      wt, idxbuf, norm_w, norm_b, Abuf);

  // (M/32) * (OUTD/64) = 288 * 8 = 2304 waves; 8 waves (256 thr) per block
  const int totalWaves = (B_ * KEEP_ / 32) * (OUTD_ / 64);
  gemm_kernel<<<totalWaves / 8, 256, 0, stream>>>(Abuf, lin_w, lin_b, feat_out);
}